// Net_46617575031249
// MI455X (gfx1250) — compile-verified
//
#include <hip/hip_runtime.h>
#include <cmath>

// MI455X / gfx1250, wave32. f32 WMMA path (V_WMMA_F32_16X16X4_F32) +
// CDNA5 async global->LDS staging (GLOBAL_LOAD_ASYNC_TO_LDS_B128, ASYNCcnt).

typedef float v2f __attribute__((ext_vector_type(2)));
typedef float v8f __attribute__((ext_vector_type(8)));

#define N_NODES 100000
#define F_IN    256
#define H       128
#define C_OUT   64
#define NLAYERS 16
#define ALPHA   0.1f

__device__ __forceinline__ v8f wmma_f32(v2f a, v2f b, v8f c) {
  // 8 args: (neg_a, A, neg_b, B, c_mod, C, reuse_a, reuse_b)
  return __builtin_amdgcn_wmma_f32_16x16x4_f32(false, a, false, b, (short)0, c,
                                               false, false);
}

// Async copy of 16 bytes global -> LDS (no VGPR round trip, ASYNCcnt-tracked).
__device__ __forceinline__ void async_g2l_b128(unsigned lds_off,
                                               const float* gaddr) {
  asm volatile("global_load_async_to_lds_b128 %0, %1, off"
               :: "v"(lds_off), "v"(gaddr) : "memory");
}
__device__ __forceinline__ void wait_asynccnt0() {
  asm volatile("s_wait_asynccnt 0x0" ::: "memory");
}

// ---------------------------------------------------------------------------
// lin0: x = relu(X[N,256] @ W[256,128] + b); write to both x_cur and x0.
// 8 waves/block, wave w computes rows [16*bid,16*bid+16) x cols [16w,16w+16).
// ---------------------------------------------------------------------------
__global__ __launch_bounds__(256) void lin0_kernel(
    const float* __restrict__ x, const float* __restrict__ w,
    const float* __restrict__ b, float* __restrict__ x_cur,
    float* __restrict__ x0) {
  const int wave = threadIdx.x >> 5;
  const int lane = threadIdx.x & 31;
  const int half = lane >> 4;
  const int l16  = lane & 15;
  const int arow = blockIdx.x * 16 + l16;
  const int col  = wave * 16 + l16;

  v8f acc = {};
#pragma unroll 4
  for (int k0 = 0; k0 < F_IN; k0 += 4) {
    const int ka = k0 + 2 * half;
    v2f a = *(const v2f*)(x + (size_t)arow * F_IN + ka);
    v2f bv;
    bv.x = w[(ka + 0) * H + col];
    bv.y = w[(ka + 1) * H + col];
    acc = wmma_f32(a, bv, acc);
  }
  const float bias = b[col];
  const int orow0 = blockIdx.x * 16 + 8 * half;
#pragma unroll
  for (int r = 0; r < 8; ++r) {
    float v = acc[r] + bias;
    v = v > 0.f ? v : 0.f;
    const int idx = (orow0 + r) * H + col;
    x_cur[idx] = v;
    x0[idx]    = v;
  }
}

// ---------------------------------------------------------------------------
// agg = ALPHA * x0   (folds the initial-residual term into the accumulator)
// ---------------------------------------------------------------------------
__global__ __launch_bounds__(256) void init_agg_kernel(
    const float* __restrict__ x0, float* __restrict__ agg, int n4) {
  int i = blockIdx.x * 256 + threadIdx.x;
  if (i < n4) {
    float4 v = ((const float4*)x0)[i];
    v.x *= ALPHA; v.y *= ALPHA; v.z *= ALPHA; v.w *= ALPHA;
    ((float4*)agg)[i] = v;
  }
}

// ---------------------------------------------------------------------------
// SpMM scatter: agg[dst] += (1-ALPHA) * w * x[src].
// One wave (32 lanes) per edge, 4 features per lane (float4 gather),
// relaxed agent-scope f32 atomics -> global_atomic_add_f32 at L2.
// ---------------------------------------------------------------------------
__global__ __launch_bounds__(256) void scatter_kernel(
    const int* __restrict__ esrc, const int* __restrict__ edst,
    const float* __restrict__ ew, const float* __restrict__ x,
    float* __restrict__ agg, int nE) {
  const int eid = blockIdx.x * 8 + (threadIdx.x >> 5);
  if (eid >= nE) return;
  const int lane = threadIdx.x & 31;
  const int s = esrc[eid];
  const int d = edst[eid];
  const float w = ew[eid] * (1.0f - ALPHA);
  const float4 v = *(const float4*)(x + (size_t)s * H + lane * 4);
  float* p = agg + (size_t)d * H + lane * 4;
  __hip_atomic_fetch_add(p + 0, w * v.x, __ATOMIC_RELAXED, __HIP_MEMORY_SCOPE_AGENT);
  __hip_atomic_fetch_add(p + 1, w * v.y, __ATOMIC_RELAXED, __HIP_MEMORY_SCOPE_AGENT);
  __hip_atomic_fetch_add(p + 2, w * v.z, __ATOMIC_RELAXED, __HIP_MEMORY_SCOPE_AGENT);
  __hip_atomic_fetch_add(p + 3, w * v.w, __ATOMIC_RELAXED, __HIP_MEMORY_SCOPE_AGENT);
}

// ---------------------------------------------------------------------------
// Fused GCN2 layer tail:  x = relu((1-beta)*h + beta*(h @ W) + x)
// h == agg (already contains alpha*x0 + (1-alpha)*aggregation).
// W (128x128 = 64KB) staged into LDS via GLOBAL_LOAD_ASYNC_TO_LDS_B128.
// In-place update of x.
// ---------------------------------------------------------------------------
__global__ __launch_bounds__(256) void conv_kernel(
    const float* __restrict__ hbuf, const float* __restrict__ wmat,
    float* __restrict__ x_io, float beta) {
  __shared__ __align__(16) float wlds[H * H];  // 64 KB of 320 KB/WGP LDS

  // Async stage W: 256 threads x 16 iters x 16B = 65536 B, VGPR-bypassing.
  for (int i = threadIdx.x * 4; i < H * H; i += 256 * 4) {
    const unsigned loff = (unsigned)(uintptr_t)(&wlds[i]);  // addr[31:0] = LDS offset
    async_g2l_b128(loff, wmat + i);
  }
  wait_asynccnt0();
  __syncthreads();

  const int wave = threadIdx.x >> 5;
  const int lane = threadIdx.x & 31;
  const int half = lane >> 4;
  const int l16  = lane & 15;
  const int arow = blockIdx.x * 16 + l16;
  const int col  = wave * 16 + l16;

  v8f acc = {};
#pragma unroll 4
  for (int k0 = 0; k0 < H; k0 += 4) {
    const int ka = k0 + 2 * half;
    v2f a = *(const v2f*)(hbuf + (size_t)arow * H + ka);
    v2f bv;
    bv.x = wlds[(ka + 0) * H + col];
    bv.y = wlds[(ka + 1) * H + col];
    acc = wmma_f32(a, bv, acc);
  }

  const float ob = 1.0f - beta;
  const int orow0 = blockIdx.x * 16 + 8 * half;
#pragma unroll
  for (int r = 0; r < 8; ++r) {
    const int idx = (orow0 + r) * H + col;
    const float hv = hbuf[idx];
    const float xv = x_io[idx];
    float v = ob * hv + beta * acc[r] + xv;
    x_io[idx] = v > 0.f ? v : 0.f;
  }
}

// ---------------------------------------------------------------------------
// lin1: out = X[N,128] @ W[128,64] + b   (4 waves/block -> 16x64 tile)
// ---------------------------------------------------------------------------
__global__ __launch_bounds__(128) void lin1_kernel(
    const float* __restrict__ x, const float* __restrict__ w,
    const float* __restrict__ b, float* __restrict__ out) {
  const int wave = threadIdx.x >> 5;
  const int lane = threadIdx.x & 31;
  const int half = lane >> 4;
  const int l16  = lane & 15;
  const int arow = blockIdx.x * 16 + l16;
  const int col  = wave * 16 + l16;

  v8f acc = {};
#pragma unroll 4
  for (int k0 = 0; k0 < H; k0 += 4) {
    const int ka = k0 + 2 * half;
    v2f a = *(const v2f*)(x + (size_t)arow * H + ka);
    v2f bv;
    bv.x = w[(ka + 0) * C_OUT + col];
    bv.y = w[(ka + 1) * C_OUT + col];
    acc = wmma_f32(a, bv, acc);
  }
  const float bias = b[col];
  const int orow0 = blockIdx.x * 16 + 8 * half;
#pragma unroll
  for (int r = 0; r < 8; ++r)
    out[(size_t)(orow0 + r) * C_OUT + col] = acc[r] + bias;
}

// ---------------------------------------------------------------------------
extern "C" void kernel_launch(void* const* d_in, const int* in_sizes, int n_in,
                              void* d_out, int out_size, void* d_ws,
                              size_t ws_size, hipStream_t stream) {
  const float* x_in   = (const float*)d_in[0];
  const int*   esrc   = (const int*)d_in[1];
  const int*   edst   = (const int*)d_in[2];
  const float* ew     = (const float*)d_in[3];
  const float* lin0_w = (const float*)d_in[4];
  const float* lin0_b = (const float*)d_in[5];
  const float* lin1_w = (const float*)d_in[6];
  const float* lin1_b = (const float*)d_in[7];
  const float* conv_w = (const float*)d_in[8];
  float* out = (float*)d_out;

  float* x_cur = (float*)d_ws;                       // N*H f32 (51.2 MB)
  float* x0    = x_cur + (size_t)N_NODES * H;        // N*H f32
  float* agg   = x0    + (size_t)N_NODES * H;        // N*H f32
  (void)ws_size; (void)n_in; (void)out_size;

  const int nE = in_sizes[1];
  const int mblocks = N_NODES / 16;                  // 6250, exact
  const int n4 = N_NODES * H / 4;

  lin0_kernel<<<mblocks, 256, 0, stream>>>(x_in, lin0_w, lin0_b, x_cur, x0);

  for (int l = 0; l < NLAYERS; ++l) {
    const float beta = logf(0.5f / (float)(l + 1) + 1.0f);
    init_agg_kernel<<<(n4 + 255) / 256, 256, 0, stream>>>(x0, agg, n4);
    scatter_kernel<<<(nE + 7) / 8, 256, 0, stream>>>(esrc, edst, ew, x_cur,
                                                     agg, nE);
    conv_kernel<<<mblocks, 256, 0, stream>>>(agg, conv_w + (size_t)l * H * H,
                                             x_cur, beta);
  }

  lin1_kernel<<<mblocks, 128, 0, stream>>>(x_cur, lin1_w, lin1_b, out);
}